// GCNBankNet_66245575573518
// MI455X (gfx1250) — compile-verified
//
#include <hip/hip_runtime.h>
#include <hip/hip_bf16.h>

typedef __attribute__((ext_vector_type(2))) float v2f;
typedef __attribute__((ext_vector_type(8))) float v8f;

// ---------------------------------------------------------------------------
// degree / norm helpers
// ---------------------------------------------------------------------------
__global__ __launch_bounds__(256) void k_zero(float* __restrict__ p, int n) {
    int i = blockIdx.x * blockDim.x + threadIdx.x;
    if (i < n) p[i] = 0.0f;
}

__global__ __launch_bounds__(256) void k_degrees(const int* __restrict__ src,
                                                 const int* __restrict__ dst,
                                                 float* __restrict__ deg_out,
                                                 float* __restrict__ deg_in,
                                                 int n_edges) {
    int e = blockIdx.x * blockDim.x + threadIdx.x;
    if (e < n_edges) {
        unsafeAtomicAdd(&deg_out[src[e]], 1.0f);  // -> global_atomic_add_f32
        unsafeAtomicAdd(&deg_in[dst[e]], 1.0f);
    }
}

__global__ __launch_bounds__(256) void k_rsqrt_clip(float* __restrict__ a,
                                                    float* __restrict__ b,
                                                    int n) {
    int i = blockIdx.x * blockDim.x + threadIdx.x;
    if (i < n) {
        a[i] = rsqrtf(fmaxf(a[i], 1.0f));
        b[i] = rsqrtf(fmaxf(b[i], 1.0f));
    }
}

// ---------------------------------------------------------------------------
// x[N,DOUT] = (h .* norm_src) @ W      h:[N,DIN] row-major, W:[DIN,DOUT] row-major
// One wave computes one 16x16 output tile with V_WMMA_F32_16X16X4_F32.
// A 16x4 layout: lane (m = lane&15, kh = lane>>4), vgpr v -> A[m][2*kh + v]
// B 4x16 layout (mirror):            vgpr v -> B[2*kh + v][lane&15]
// C/D 16x16:  vgpr r, lane l -> D[r + 8*(l>>4)][l&15]
// Requires n_nodes % 16 == 0 (50000 = 3125*16).
// ---------------------------------------------------------------------------
template<int DIN, int DOUT>
__global__ __launch_bounds__(256)
void gcn_gemm_wmma(const float* __restrict__ h,
                   const float* __restrict__ norm_src,
                   const float* __restrict__ W,
                   float* __restrict__ x,
                   int n_nodes) {
    const int lane = threadIdx.x & 31;
    const int wave = threadIdx.x >> 5;
    const int tiles_n = DOUT / 16;

    const int tile   = blockIdx.x * 8 + wave;     // 8 waves per block
    const int tile_m = tile / tiles_n;
    const int tile_n = tile % tiles_n;
    if (tile_m * 16 >= n_nodes) return;           // uniform per wave

    const int m  = lane & 15;                     // row within A-tile / col within B,D-tile
    const int kh = lane >> 4;                     // K half-select

    const int   row  = tile_m * 16 + m;
    const int   col  = tile_n * 16 + m;
    const float ns   = norm_src[row];
    const float* __restrict__ hrow = h + (size_t)row * DIN;
    const float* __restrict__ wcol = W + col;

    v8f c = {};
#pragma unroll
    for (int k0 = 0; k0 < DIN; k0 += 4) {
        const int k = k0 + 2 * kh;
        v2f a, b;
        a.x = hrow[k]     * ns;                   // contiguous -> b64 load
        a.y = hrow[k + 1] * ns;
        b.x = wcol[(size_t)k * DOUT];
        b.y = wcol[(size_t)(k + 1) * DOUT];
        c = __builtin_amdgcn_wmma_f32_16x16x4_f32(false, a, false, b,
                                                  (short)0, c, false, false);
    }

    float* __restrict__ xout =
        x + (size_t)(tile_m * 16 + 8 * kh) * DOUT + tile_n * 16 + m;
#pragma unroll
    for (int r = 0; r < 8; ++r)
        xout[(size_t)r * DOUT] = c[r];
}

// ---------------------------------------------------------------------------
// agg[dst[e]] += x[src[e]]   (float4 per thread, hardware f32 atomics)
// ---------------------------------------------------------------------------
template<int DOUT>
__global__ __launch_bounds__(256)
void k_scatter(const float* __restrict__ x,
               const int* __restrict__ src,
               const int* __restrict__ dst,
               float* __restrict__ agg,
               int n_edges) {
    constexpr int CHUNKS = DOUT / 4;
    long long gid = (long long)blockIdx.x * blockDim.x + threadIdx.x;
    if (gid >= (long long)n_edges * CHUNKS) return;
    const int e  = (int)(gid / CHUNKS);
    const int cw = (int)(gid % CHUNKS);
    const int s = src[e];
    const int d = dst[e];
    const float4 v = *(const float4*)(x + (size_t)s * DOUT + cw * 4);
    float* out = agg + (size_t)d * DOUT + cw * 4;
    unsafeAtomicAdd(out + 0, v.x);
    unsafeAtomicAdd(out + 1, v.y);
    unsafeAtomicAdd(out + 2, v.z);
    unsafeAtomicAdd(out + 3, v.w);
}

// ---------------------------------------------------------------------------
// out = (RELU?) (agg * norm_dst + b)
// ---------------------------------------------------------------------------
template<int DOUT, bool RELU>
__global__ __launch_bounds__(256)
void k_finalize(const float* __restrict__ agg,
                const float* __restrict__ norm_dst,
                const float* __restrict__ bias,
                float* __restrict__ out,
                int n_nodes) {
    int gid = blockIdx.x * blockDim.x + threadIdx.x;
    if (gid >= n_nodes * DOUT) return;
    const int i = gid / DOUT;
    const int j = gid % DOUT;
    float v = agg[gid] * norm_dst[i] + bias[j];
    if (RELU) v = fmaxf(v, 0.0f);
    out[gid] = v;
}

// ---------------------------------------------------------------------------
extern "C" void kernel_launch(void* const* d_in, const int* in_sizes, int n_in,
                              void* d_out, int out_size, void* d_ws, size_t ws_size,
                              hipStream_t stream) {
    constexpr int DIN = 128, HID = 128, DOUT_LAST = 64;

    const float* h0  = (const float*)d_in[0];
    const int*   src = (const int*)d_in[1];
    const int*   dst = (const int*)d_in[2];
    const float* Wm[4] = {(const float*)d_in[3], (const float*)d_in[5],
                          (const float*)d_in[7], (const float*)d_in[9]};
    const float* Bv[4] = {(const float*)d_in[4], (const float*)d_in[6],
                          (const float*)d_in[8], (const float*)d_in[10]};

    const int N = in_sizes[0] / DIN;   // 50000 (multiple of 16)
    const int E = in_sizes[1];         // 600000

    // workspace layout (floats): norm_src[N] | norm_dst[N] | cur[N*HID] | x[N*HID] | agg[N*HID]
    float* ws       = (float*)d_ws;
    float* norm_src = ws;
    float* norm_dst = ws + N;
    float* cur      = ws + 2 * (size_t)N;
    float* xbuf     = cur  + (size_t)N * HID;
    float* agg      = xbuf + (size_t)N * HID;

    auto cdiv = [](long long a, long long b) { return (int)((a + b - 1) / b); };

    // --- degree norms ---
    k_zero<<<cdiv(2LL * N, 256), 256, 0, stream>>>(norm_src, 2 * N); // contiguous pair
    k_degrees<<<cdiv(E, 256), 256, 0, stream>>>(src, dst, norm_src, norm_dst, E);
    k_rsqrt_clip<<<cdiv(N, 256), 256, 0, stream>>>(norm_src, norm_dst, N);

    // --- layers ---
    const float* hin = h0;
    for (int L = 0; L < 4; ++L) {
        const int dout  = (L == 3) ? DOUT_LAST : HID;
        const int tiles = (N / 16) * (dout / 16);

        if (dout == HID)
            gcn_gemm_wmma<DIN, HID><<<cdiv(tiles, 8), 256, 0, stream>>>(
                hin, norm_src, Wm[L], xbuf, N);
        else
            gcn_gemm_wmma<DIN, DOUT_LAST><<<cdiv(tiles, 8), 256, 0, stream>>>(
                hin, norm_src, Wm[L], xbuf, N);

        k_zero<<<cdiv((long long)N * dout, 256), 256, 0, stream>>>(agg, N * dout);

        if (dout == HID)
            k_scatter<HID><<<cdiv((long long)E * (HID / 4), 256), 256, 0, stream>>>(
                xbuf, src, dst, agg, E);
        else
            k_scatter<DOUT_LAST><<<cdiv((long long)E * (DOUT_LAST / 4), 256), 256, 0, stream>>>(
                xbuf, src, dst, agg, E);

        if (L == 3)
            k_finalize<DOUT_LAST, false><<<cdiv((long long)N * DOUT_LAST, 256), 256, 0, stream>>>(
                agg, norm_dst, Bv[L], (float*)d_out, N);
        else
            k_finalize<HID, true><<<cdiv((long long)N * HID, 256), 256, 0, stream>>>(
                agg, norm_dst, Bv[L], cur, N);

        hin = cur;
    }
}